// Adaptive_Wavelet_Filter_Block_65352222376679
// MI455X (gfx1250) — compile-verified
//
#include <hip/hip_runtime.h>
#include <math.h>

typedef __attribute__((ext_vector_type(2))) float v2f;
typedef __attribute__((ext_vector_type(8))) float v8f;

// Problem dims
#define BDIM 8
#define NDIM 4096
#define CDIM 1024
#define L0 2048
#define L1 1024
#define L2 512
#define L3 256

// Energy-row bases (scale3 detail is provably unused by the reference)
#define ROW1_BASE (BDIM * L0)              // 16384
#define ROW2_BASE (ROW1_BASE + BDIM * L1)  // 24576
#define NROWS (ROW2_BASE + BDIM * L2)      // 28672  (divisible by 16)

// Workspace layout (in floats)
static constexpr size_t OFF_D0 = 0;
static constexpr size_t OFF_D1 = OFF_D0 + (size_t)BDIM * L0 * CDIM;   // 16,777,216
static constexpr size_t OFF_D2 = OFF_D1 + (size_t)BDIM * L1 * CDIM;   // 25,165,824
static constexpr size_t OFF_A3 = OFF_D2 + (size_t)BDIM * L2 * CDIM;   // 29,360,128
static constexpr size_t OFF_REC1 = OFF_A3 + (size_t)BDIM * L3 * CDIM; // 31,457,280
static constexpr size_t OFF_PART = OFF_REC1 + (size_t)BDIM * L1 * CDIM; // 39,845,888
static constexpr size_t OFF_EN = OFF_PART + (size_t)NROWS * 32;       // 40,763,392
static constexpr size_t OFF_SC = OFF_EN + (size_t)NROWS;              // 40,792,064
// scalars: [0]=mix  [1..4]=sw0..sw3  [5..7]=thr0..thr2

// ---------------------------------------------------------------- K1: scalars
__global__ __launch_bounds__(256) void k_scalars(
    const float* __restrict__ cw1, const float* __restrict__ cw2,
    const float* __restrict__ cw3, const float* __restrict__ swt,
    const float* __restrict__ rw, float* __restrict__ sc) {
  __shared__ float s1[256], s2[256], s3[256];
  int t = threadIdx.x;
  float a = 0.f, b = 0.f, c = 0.f;
  for (int i = t; i < CDIM; i += 256) { a += cw1[i]; b += cw2[i]; c += cw3[i]; }
  s1[t] = a; s2[t] = b; s3[t] = c;
  __syncthreads();
  for (int off = 128; off > 0; off >>= 1) {
    if (t < off) { s1[t] += s1[t + off]; s2[t] += s2[t + off]; s3[t] += s3[t + off]; }
    __syncthreads();
  }
  if (t == 0) {
    const float inv = 1.0f / (float)CDIM;
    float m1 = s1[0] * inv, m2 = s2[0] * inv, m3 = s3[0] * inv;
    float sig = 1.0f / (1.0f + expf(-rw[0]));
    sc[0] = (m1 + 0.5f * m2 + 0.2f * m3) * sig;
    for (int i = 0; i < 4; i++) sc[1 + i] = 1.0f / (1.0f + expf(-swt[i]));
  }
}

// ------------------------------------------------------- K2: Haar + partials
__device__ __forceinline__ void wave_partial(float v, int rowid, float* part) {
  v += __shfl_xor(v, 1);
  v += __shfl_xor(v, 2);
  v += __shfl_xor(v, 4);
  v += __shfl_xor(v, 8);
  v += __shfl_xor(v, 16);
  if ((threadIdx.x & 31) == 0)
    part[(size_t)rowid * 32 + (threadIdx.x >> 5)] = v;
}

__global__ __launch_bounds__(1024) void k_haar(const float* __restrict__ x,
                                               float* __restrict__ ws) {
  float* d0 = ws + OFF_D0;
  float* d1 = ws + OFF_D1;
  float* d2 = ws + OFF_D2;
  float* a3 = ws + OFF_A3;
  float* part = ws + OFF_PART;

  int b = blockIdx.x >> 6;       // 8 batches
  int chunk = blockIdx.x & 63;   // 64 n-chunks of 64
  int n0 = chunk * 64;
  int c = threadIdx.x;

  const float* xb = x + ((size_t)b * NDIM + n0) * CDIM + c;
  float c0 = 0.f, c1 = 0.f, c2 = 0.f;

#pragma unroll
  for (int i = 0; i < 32; i++) {  // 32 scale-0 pairs in this chunk
    // x is read exactly once across the whole pipeline -> bypass L2 residency
    float x0 = __builtin_nontemporal_load(&xb[(size_t)(2 * i) * CDIM]);
    float x1 = __builtin_nontemporal_load(&xb[(size_t)(2 * i + 1) * CDIM]);
    float a0 = (x0 + x1) * 0.5f;
    float dv = (x0 - x1) * 0.5f;
    int l0 = (n0 >> 1) + i;
    d0[((size_t)b * L0 + l0) * CDIM + c] = dv;  // re-read later: keep cached
    wave_partial(dv * dv, b * L0 + l0, part);
    if ((i & 1) == 0) {
      c0 = a0;
    } else {
      float a1 = (c0 + a0) * 0.5f, d1v = (c0 - a0) * 0.5f;
      int l1 = (n0 >> 2) + (i >> 1);
      d1[((size_t)b * L1 + l1) * CDIM + c] = d1v;
      wave_partial(d1v * d1v, ROW1_BASE + b * L1 + l1, part);
      if ((i & 3) == 1) {
        c1 = a1;
      } else {
        float a2 = (c1 + a1) * 0.5f, d2v = (c1 - a1) * 0.5f;
        int l2 = (n0 >> 3) + (i >> 2);
        d2[((size_t)b * L2 + l2) * CDIM + c] = d2v;
        wave_partial(d2v * d2v, ROW2_BASE + b * L2 + l2, part);
        if ((i & 7) == 3) {
          c2 = a2;
        } else {
          float a3v = (c2 + a2) * 0.5f;  // detail3 unused by reference
          int l3 = (n0 >> 4) + (i >> 3);
          a3[((size_t)b * L3 + l3) * CDIM + c] = a3v;
        }
      }
    }
  }
}

// ------------------------------------ K3: energy row-sums on the matrix pipe
// partial is NROWS x 32; D = A(16x4) * ones(4x16) + C accumulates row sums.
__global__ __launch_bounds__(256) void k_energy(const float* __restrict__ part,
                                                float* __restrict__ energy) {
  int wave = blockIdx.x * 8 + (threadIdx.x >> 5);  // 1792 waves total
  int lane = threadIdx.x & 31;
  int r0 = wave * 16;
  int row = r0 + (lane & 15);
  int col0 = (lane >> 4) << 1;  // lanes 16..31 hold K=2,3
  const float* basep = part + (size_t)row * 32 + col0;

  v8f acc = {0.f, 0.f, 0.f, 0.f, 0.f, 0.f, 0.f, 0.f};
  v2f bones; bones[0] = 1.0f; bones[1] = 1.0f;

#pragma unroll
  for (int k = 0; k < 8; k++) {
    v2f a;
    a[0] = __builtin_nontemporal_load(&basep[k * 4 + 0]);
    a[1] = __builtin_nontemporal_load(&basep[k * 4 + 1]);
    acc = __builtin_amdgcn_wmma_f32_16x16x4_f32(false, a, false, bones,
                                                (short)0, acc, false, false);
  }

  // C/D layout: VGPR j = {M=j @ lanes 0-15, M=j+8 @ lanes 16-31}, N = lane%16
  if (lane == 0 || lane == 16) {
    int rb = r0 + ((lane >> 4) << 3);
    const float inv = 1.0f / (float)CDIM;
    energy[rb + 0] = acc[0] * inv;
    energy[rb + 1] = acc[1] * inv;
    energy[rb + 2] = acc[2] * inv;
    energy[rb + 3] = acc[3] * inv;
    energy[rb + 4] = acc[4] * inv;
    energy[rb + 5] = acc[5] * inv;
    energy[rb + 6] = acc[6] * inv;
    energy[rb + 7] = acc[7] * inv;
  }
}

// ----------------------------------------- K4: bitonic sort + quantile thr
__global__ __launch_bounds__(1024) void k_sortthr(const float* __restrict__ energy,
                                                  const float* __restrict__ thrp,
                                                  float* __restrict__ sc) {
  __shared__ float sm[16384];
  int s = blockIdx.x;                 // scale 0..2
  int n = 16384 >> s;                 // 16384, 8192, 4096
  int base = (s == 0) ? 0 : ((s == 1) ? ROW1_BASE : ROW2_BASE);
  const float* e = energy + base;
  for (int i = threadIdx.x; i < n; i += 1024) sm[i] = e[i];
  __syncthreads();

  for (int k = 2; k <= n; k <<= 1) {
    for (int j = k >> 1; j > 0; j >>= 1) {
      for (int i = threadIdx.x; i < n; i += 1024) {
        int ix = i ^ j;
        if (ix > i) {
          float a = sm[i], b = sm[ix];
          bool up = ((i & k) == 0);
          if ((a > b) == up) { sm[i] = b; sm[ix] = a; }
        }
      }
      __syncthreads();
    }
  }

  if (threadIdx.x == 0) {
    float q = thrp[0];
    float idxf = q * (float)(n - 1);
    int lo = (int)floorf(idxf);
    if (lo < 0) lo = 0;
    if (lo > n - 1) lo = n - 1;
    int hi = lo + 1;
    if (hi > n - 1) hi = n - 1;
    float frac = idxf - (float)lo;
    sc[5 + s] = sm[lo] * (1.0f - frac) + sm[hi] * frac;
  }
}

// ---------------------------------------------- K5a: rec1 = up(up(a3s)+d2m)+d1m
__device__ __forceinline__ float a3val(const float* a3, int b, int l3, int c,
                                       float sw3) {
  return a3[((size_t)b * L3 + l3) * CDIM + c] * sw3;
}

__device__ __forceinline__ float rec2val(const float* a3, const float* d2,
                                         const float* en, int b, int l2, int c,
                                         float sw3, float sw2, float thr2) {
  int j = l2 >> 1;
  float up;
  if ((l2 & 1) == 0) {
    int jm = j - 1; if (jm < 0) jm = 0;
    up = 0.25f * a3val(a3, b, jm, c, sw3) + 0.75f * a3val(a3, b, j, c, sw3);
  } else {
    int jp = j + 1; if (jp > L3 - 1) jp = L3 - 1;
    up = 0.75f * a3val(a3, b, j, c, sw3) + 0.25f * a3val(a3, b, jp, c, sw3);
  }
  float e = en[ROW2_BASE + b * L2 + l2];
  float dm = (e > thr2) ? d2[((size_t)b * L2 + l2) * CDIM + c] * sw2 : 0.f;
  return up + dm;
}

__global__ __launch_bounds__(256) void k_rec1(float* __restrict__ ws) {
  const float* sc = ws + OFF_SC;
  size_t idx = (size_t)blockIdx.x * 256 + threadIdx.x;  // 8,388,608 total
  int c = (int)(idx & (CDIM - 1));
  size_t t = idx >> 10;
  int l1 = (int)(t & (L1 - 1));
  int b = (int)(t >> 10);
  float sw1 = sc[2], sw2 = sc[3], sw3 = sc[4], thr1 = sc[6], thr2 = sc[7];
  const float* a3 = ws + OFF_A3;
  const float* d2 = ws + OFF_D2;
  const float* d1 = ws + OFF_D1;
  const float* en = ws + OFF_EN;

  int j = l1 >> 1;
  float up;
  if ((l1 & 1) == 0) {
    int jm = j - 1; if (jm < 0) jm = 0;
    up = 0.25f * rec2val(a3, d2, en, b, jm, c, sw3, sw2, thr2) +
         0.75f * rec2val(a3, d2, en, b, j, c, sw3, sw2, thr2);
  } else {
    int jp = j + 1; if (jp > L2 - 1) jp = L2 - 1;
    up = 0.75f * rec2val(a3, d2, en, b, j, c, sw3, sw2, thr2) +
         0.25f * rec2val(a3, d2, en, b, jp, c, sw3, sw2, thr2);
  }
  float e = en[ROW1_BASE + b * L1 + l1];
  float dm = (e > thr1) ? d1[idx] * sw1 : 0.f;
  ws[OFF_REC1 + idx] = up + dm;  // re-read by k_out: keep cached
}

// ------------------------------------------------ K5b: out = up(up(rec1)+d0m)*mix
__device__ __forceinline__ float rec0val(const float* rec1, const float* d0,
                                         const float* en, int b, int l0, int c,
                                         float sw0, float thr0) {
  int j = l0 >> 1;
  float up;
  if ((l0 & 1) == 0) {
    int jm = j - 1; if (jm < 0) jm = 0;
    up = 0.25f * rec1[((size_t)b * L1 + jm) * CDIM + c] +
         0.75f * rec1[((size_t)b * L1 + j) * CDIM + c];
  } else {
    int jp = j + 1; if (jp > L1 - 1) jp = L1 - 1;
    up = 0.75f * rec1[((size_t)b * L1 + j) * CDIM + c] +
         0.25f * rec1[((size_t)b * L1 + jp) * CDIM + c];
  }
  float e = en[b * L0 + l0];
  float dm = (e > thr0) ? d0[((size_t)b * L0 + l0) * CDIM + c] * sw0 : 0.f;
  return up + dm;
}

__global__ __launch_bounds__(256) void k_out(const float* __restrict__ ws,
                                             float* __restrict__ out) {
  const float* sc = ws + OFF_SC;
  size_t idx = (size_t)blockIdx.x * 256 + threadIdx.x;  // 33,554,432 total
  int c = (int)(idx & (CDIM - 1));
  size_t t = idx >> 10;
  int n = (int)(t & (NDIM - 1));
  int b = (int)(t >> 12);
  float mix = sc[0], sw0 = sc[1], thr0 = sc[5];
  const float* rec1 = ws + OFF_REC1;
  const float* d0 = ws + OFF_D0;
  const float* en = ws + OFF_EN;

  int j = n >> 1;
  float v;
  if ((n & 1) == 0) {
    int jm = j - 1; if (jm < 0) jm = 0;
    v = 0.25f * rec0val(rec1, d0, en, b, jm, c, sw0, thr0) +
        0.75f * rec0val(rec1, d0, en, b, j, c, sw0, thr0);
  } else {
    int jp = j + 1; if (jp > L0 - 1) jp = L0 - 1;
    v = 0.75f * rec0val(rec1, d0, en, b, j, c, sw0, thr0) +
        0.25f * rec0val(rec1, d0, en, b, jp, c, sw0, thr0);
  }
  // out is never re-read: stream it past L2 so d0/rec1 stay resident
  __builtin_nontemporal_store(v * mix, &out[idx]);
}

// ----------------------------------------------------------------- launcher
extern "C" void kernel_launch(void* const* d_in, const int* in_sizes, int n_in,
                              void* d_out, int out_size, void* d_ws,
                              size_t ws_size, hipStream_t stream) {
  const float* x = (const float*)d_in[0];
  const float* cw1 = (const float*)d_in[1];
  const float* cw2 = (const float*)d_in[2];
  const float* cw3 = (const float*)d_in[3];
  const float* thrp = (const float*)d_in[4];
  const float* swt = (const float*)d_in[5];
  const float* rw = (const float*)d_in[6];
  float* ws = (float*)d_ws;
  float* out = (float*)d_out;

  k_scalars<<<1, 256, 0, stream>>>(cw1, cw2, cw3, swt, rw, ws + OFF_SC);
  k_haar<<<BDIM * 64, 1024, 0, stream>>>(x, ws);
  k_energy<<<(NROWS / 16) / 8, 256, 0, stream>>>(ws + OFF_PART, ws + OFF_EN);
  k_sortthr<<<3, 1024, 0, stream>>>(ws + OFF_EN, thrp, ws + OFF_SC);
  k_rec1<<<(BDIM * L1 * CDIM) / 256, 256, 0, stream>>>(ws);
  k_out<<<(BDIM * NDIM * CDIM) / 256, 256, 0, stream>>>(ws, out);
}